// ConstrainedAttentionModel_41798621724914
// MI455X (gfx1250) — compile-verified
//
#include <hip/hip_runtime.h>
#include <hip/hip_bf16.h>
#include <math.h>

#define VOCAB 8192
#define TLEN  2048
#define NTHR  256
#define PER_T (TLEN / NTHR)   // 8 score elements per thread

#if __has_builtin(__builtin_amdgcn_wmma_f32_16x16x4_f32)
#define USE_WMMA 1
#else
#define USE_WMMA 0
#endif

typedef float v2f __attribute__((ext_vector_type(2)));
typedef float v8f __attribute__((ext_vector_type(8)));

__global__ __launch_bounds__(NTHR) void constrained_attn_kernel(
    const int* __restrict__ x,        // (8, 2048) int32
    const float* __restrict__ params, // (3, 3) f32
    float* __restrict__ out)          // (8, 8192) f32
{
    __shared__ __align__(16) int   xs[TLEN];    // 8 KB token row
    __shared__ float               hist[VOCAB]; // 32 KB weighted histogram
    __shared__ float               red[NTHR / 32];
#if USE_WMMA
    __shared__ float               dcol[TLEN * 4]; // 32 KB: D columns 0..2 (stride 4)
#endif

    const int tid  = threadIdx.x;
    const int b    = blockIdx.x;
    const int* xrow = x + b * TLEN;

    // ---- Stage the 8 KB token row into LDS with CDNA5 async global->LDS
    // loads (ASYNCcnt path, bypasses VGPRs). 256 lanes x 16 B x 2 = 8 KB.
    {
        unsigned lds0          = (unsigned)(size_t)(&xs[0]) + (unsigned)tid * 16u;
        unsigned long long ga0 = (unsigned long long)(size_t)(xrow + tid * 4);
        asm volatile("global_load_async_to_lds_b128 %0, %1, off"
                     :: "v"(lds0), "v"(ga0) : "memory");
        unsigned lds1          = lds0 + 4096u;
        unsigned long long ga1 = ga0 + 4096ull;
        asm volatile("global_load_async_to_lds_b128 %0, %1, off"
                     :: "v"(lds1), "v"(ga1) : "memory");
    }

    // ---- Zero the LDS histogram while the async loads are in flight.
    for (int v = tid; v < VOCAB; v += NTHR) hist[v] = 0.0f;

    // ---- params: uniform address -> scalar loads.
    float p[9];
#pragma unroll
    for (int k = 0; k < 9; ++k) p[k] = params[k];

    asm volatile("s_wait_asynccnt 0" ::: "memory");
    __syncthreads();

    const int q0 = xs[TLEN - 1];  // x[b, T-1]  (i = 0)
    const int q1 = xs[TLEN - 2];  // x[b, T-2]  (i = 1)
    const int q2 = xs[TLEN - 3];  // x[b, T-3]  (i = 2)

#if USE_WMMA
    // ==== Phase A: D = M x P via V_WMMA_F32_16X16X4_F32.
    //   M (Tx4): M[t][k] = [x[t]==q_k] (k=3 pad=0);  P (4x16): P[k][n]=params[k][n], n<3.
    //   D[t][n] = sum_k m_k(t)*params[k][n]  -> columns 0..2 spilled to LDS.
    {
        const int lane = tid & 31;
        const int half = lane >> 4;   // 0: K=0,1 / rows M=0..7 of D; 1: K=2,3 / rows M=8..15
        const int sub  = lane & 15;
        const int wid  = tid >> 5;    // 8 waves

        // B-matrix (4x16): lane n holds column n; VGPR0 = rows k0/k2, VGPR1 = k1/k3.
        v2f bmat;
        bmat.x = 0.0f; bmat.y = 0.0f;
        if (sub < 3) {
            bmat.x = half ? p[2 * 3 + sub] : p[0 * 3 + sub];  // k=2 : k=0
            bmat.y = half ? 0.0f           : p[1 * 3 + sub];  // k=3 : k=1
        }

        for (int g = wid; g < TLEN / 16; g += NTHR / 32) {
            const int base = g * 16;
            const int xv   = xs[base + sub];  // A row = token base+sub (both halves)
            v2f a;
            if (half == 0) {
                a.x = (xv == q0) ? 1.0f : 0.0f;   // K=0
                a.y = (xv == q1) ? 1.0f : 0.0f;   // K=1
            } else {
                a.x = (xv == q2) ? 1.0f : 0.0f;   // K=2
                a.y = 0.0f;                       // K=3 (pad)
            }
            v8f c = {};
            c = __builtin_amdgcn_wmma_f32_16x16x4_f32(
                    /*neg_a=*/false, a, /*neg_b=*/false, bmat,
                    /*c_mod=*/(short)0, c, /*reuse_a=*/false, /*reuse_b=*/false);
            // D layout: VGPR r -> M = r + 8*half, N = sub. Keep columns 0..2.
            if (sub < 3) {
                const int rowbase = base + half * 8;
#pragma unroll
                for (int r = 0; r < 8; ++r)
                    dcol[(rowbase + r) * 4 + sub] = c[r];
            }
        }
    }
    __syncthreads();
#endif

    // ==== Phase B: assemble scores, softmax, scatter.
    float sc[PER_T];
    float lmax = -INFINITY;
#pragma unroll
    for (int r = 0; r < PER_T; ++r) {
        const int t = tid + r * NTHR;
        float s;
#if USE_WMMA
        s = dcol[t * 4 + 0];                        // j=0
        if (t >= 1) s += dcol[(t - 1) * 4 + 1];     // j=1
        if (t >= 2) s += dcol[(t - 2) * 4 + 2];     // j=2
#else
        s = 0.0f;
#pragma unroll
        for (int j = 0; j < 3; ++j) {
            const int tj = t - j;
            if (tj >= 0) {
                const int xv = xs[tj];
                if (xv == q0) s += p[0 * 3 + j];
                if (xv == q1) s += p[1 * 3 + j];
                if (xv == q2) s += p[2 * 3 + j];
            }
        }
#endif
        if (t == TLEN - 1) s = -1.0e9f;  // mask last position
        sc[r] = s;
        lmax = fmaxf(lmax, s);
    }

    // ---- Block max (wave32 shuffle tree + 8-wave LDS combine).
#pragma unroll
    for (int off = 16; off > 0; off >>= 1)
        lmax = fmaxf(lmax, __shfl_xor(lmax, off, 32));
    if ((tid & 31) == 0) red[tid >> 5] = lmax;
    __syncthreads();
    float bmax = red[0];
#pragma unroll
    for (int w = 1; w < NTHR / 32; ++w) bmax = fmaxf(bmax, red[w]);
    __syncthreads();  // red[] reused below

    // ---- exp + block sum.
    float lsum = 0.0f;
#pragma unroll
    for (int r = 0; r < PER_T; ++r) {
        const float e = expf(sc[r] - bmax);
        sc[r] = e;
        lsum += e;
    }
#pragma unroll
    for (int off = 16; off > 0; off >>= 1)
        lsum += __shfl_xor(lsum, off, 32);
    if ((tid & 31) == 0) red[tid >> 5] = lsum;
    __syncthreads();
    float bsum = 0.0f;
#pragma unroll
    for (int w = 0; w < NTHR / 32; ++w) bsum += red[w];
    const float inv = 1.0f / bsum;

    // ---- Weighted scatter into LDS histogram (ds_add_f32 atomics).
#pragma unroll
    for (int r = 0; r < PER_T; ++r) {
        const int t = tid + r * NTHR;
        atomicAdd(&hist[xs[t]], sc[r] * inv);  // masked slot adds ~0
    }
    __syncthreads();

    // ---- Coalesced write-out of the full vocab row.
    float* orow = out + b * VOCAB;
    for (int v = tid; v < VOCAB; v += NTHR) orow[v] = hist[v];
}

extern "C" void kernel_launch(void* const* d_in, const int* in_sizes, int n_in,
                              void* d_out, int out_size, void* d_ws, size_t ws_size,
                              hipStream_t stream) {
    const int*   x      = (const int*)d_in[0];    // (8, 2048) int32
    const float* params = (const float*)d_in[1];  // (3, 3) f32
    float*       out    = (float*)d_out;          // (8, 8192) f32

    constrained_attn_kernel<<<dim3(8), dim3(NTHR), 0, stream>>>(x, params, out);
}